// FIRFilter1DLinearPhaseI_29953101922707
// MI455X (gfx1250) — compile-verified
//
#include <hip/hip_runtime.h>

// FIR (255-tap, linear phase type I) full convolution on (16,2,262144) fp32,
// as banded-Toeplitz GEMM on the CDNA5 matrix unit:
//   y_m[p+n] = sum_k A[m,k]*B[k,n],  A[m,k]=x_m[p-254+k],  B[k,n]=h[254+n-k]
// using V_WMMA_F32_16X16X4_F32 (full fp32 precision, 2048 FLOP/instr).
// K loop outermost per wave: one B fragment feeds 4 accumulator tiles
// (4 independent WMMAs -> no D->A/B hazards, 4x fewer B fetches).

typedef float v2f __attribute__((ext_vector_type(2)));
typedef float v8f __attribute__((ext_vector_type(8)));

#define N_IN    262144
#define N_TAPS  255
#define M_OUT   (N_IN + N_TAPS - 1)     // 262398
#define OUT_TILE 512                    // output positions per workgroup
#define KBLK    68                      // ceil(270/4): K window rounded to 272
#define LDX     770                     // LDS row stride (floats): even -> 8B-aligned b64 reads
#define HR_SIZE 288                     // reversed padded taps: hr[c] = h(270-c), zero outside

__global__ __launch_bounds__(256)
void fir_wmma_f32_kernel(const float* __restrict__ x,
                         const float* __restrict__ taps,
                         float* __restrict__ y)
{
    __shared__ float xt[16 * LDX];      // 16 rows x 768 (+pad) window of x
    __shared__ float hr[HR_SIZE];       // reversed zero-padded symmetric taps

    const int  tid     = threadIdx.x;
    const int  rg      = blockIdx.y;                     // row group: rows rg*16 .. rg*16+15
    const long wg_pout = (long)blockIdx.x * OUT_TILE;    // first output position of tile
    const long wg_p0   = wg_pout - (N_TAPS - 1);         // first x position needed

    // hr[c] = h(270-c); h(j) = taps[|j-127|] for j in [0,255), else 0.
    for (int c = tid; c < HR_SIZE; c += 256) {
        int j = 270 - c;
        float v = 0.f;
        if (j >= 0 && j < N_TAPS) {
            int t = j - 127; if (t < 0) t = -t;
            v = taps[t];
        }
        hr[c] = v;
    }

    // Stage x tile into LDS: 16 rows x 768 floats, zero-padded, float4 global loads.
    for (int idx = tid; idx < 16 * 192; idx += 256) {
        const int  row = idx / 192;
        const int  c4  = (idx % 192) * 4;
        const long g   = wg_p0 + c4;
        const float* __restrict__ src = x + (long)(rg * 16 + row) * N_IN;
        float4 v;
        if (g >= 0 && g + 3 < (long)N_IN) {
            v = *(const float4*)(src + g);
        } else {
            v.x = (g + 0 >= 0 && g + 0 < (long)N_IN) ? src[g + 0] : 0.f;
            v.y = (g + 1 >= 0 && g + 1 < (long)N_IN) ? src[g + 1] : 0.f;
            v.z = (g + 2 >= 0 && g + 2 < (long)N_IN) ? src[g + 2] : 0.f;
            v.w = (g + 3 >= 0 && g + 3 < (long)N_IN) ? src[g + 3] : 0.f;
        }
        float* d = &xt[row * LDX + c4];
        d[0] = v.x; d[1] = v.y; d[2] = v.z; d[3] = v.w;
    }
    __syncthreads();

    const int wave = tid >> 5;          // 0..7: positions wave*64 .. wave*64+63
    const int lane = tid & 31;
    const int mn   = lane & 15;         // A: row m; B/C: column n
    const int kp   = lane >> 4;         // half-wave K offset (0 or 1)

    // A fragment base: xt[m*LDX + wave*64 + blk*16 + 2*kp + k0 + {0,1}]
    const float* __restrict__ arow = &xt[mn * LDX + wave * 64 + 2 * kp];
    // B fragment base (ascending pair): hr[rb0 + k0], hr[rb0 + k0 + 1]
    //   == h(254 + n - (k0 + 2*kp)), h(254 + n - (k0 + 2*kp) - 1)
    const int rb0 = 16 - mn + 2 * kp;

    v8f acc0 = {0.f,0.f,0.f,0.f,0.f,0.f,0.f,0.f};
    v8f acc1 = acc0, acc2 = acc0, acc3 = acc0;

    #pragma unroll 2
    for (int kb = 0; kb < KBLK; ++kb) {
        const int k0 = kb * 4;
        v2f b;
        b.x = hr[rb0 + k0];
        b.y = hr[rb0 + k0 + 1];
        v2f a0 = *(const v2f*)(arow + k0);        // ds_load_b64 (8B aligned)
        v2f a1 = *(const v2f*)(arow + k0 + 16);
        v2f a2 = *(const v2f*)(arow + k0 + 32);
        v2f a3 = *(const v2f*)(arow + k0 + 48);
        acc0 = __builtin_amdgcn_wmma_f32_16x16x4_f32(false, a0, false, b, (short)0, acc0, false, false);
        acc1 = __builtin_amdgcn_wmma_f32_16x16x4_f32(false, a1, false, b, (short)0, acc1, false, false);
        acc2 = __builtin_amdgcn_wmma_f32_16x16x4_f32(false, a2, false, b, (short)0, acc2, false, false);
        acc3 = __builtin_amdgcn_wmma_f32_16x16x4_f32(false, a3, false, b, (short)0, acc3, false, false);
    }

    // C layout: VGPR v -> row (v + 8*kp), col n. Coalesced 16-float row bursts.
    const long rowbase = (long)(rg * 16) * M_OUT;
    #pragma unroll
    for (int blk = 0; blk < 4; ++blk) {
        const v8f acc = (blk == 0) ? acc0 : (blk == 1) ? acc1 : (blk == 2) ? acc2 : acc3;
        const long pout = wg_pout + wave * 64 + blk * 16 + mn;
        if (pout < (long)M_OUT) {
            #pragma unroll
            for (int v = 0; v < 8; ++v) {
                const int row = v + 8 * kp;
                y[rowbase + (long)row * M_OUT + pout] = acc[v];
            }
        }
    }
}

extern "C" void kernel_launch(void* const* d_in, const int* in_sizes, int n_in,
                              void* d_out, int out_size, void* d_ws, size_t ws_size,
                              hipStream_t stream) {
    const float* x    = (const float*)d_in[0];   // (16, 2, 262144) fp32
    const float* taps = (const float*)d_in[1];   // (128, 1) fp32
    float* y          = (float*)d_out;           // (16, 2, 262398) fp32

    dim3 grid((M_OUT + OUT_TILE - 1) / OUT_TILE, /*row groups=*/2, 1);
    dim3 block(256, 1, 1);
    fir_wmma_f32_kernel<<<grid, block, 0, stream>>>(x, taps, y);
}